// DepthBBoxProcessor_2645699854492
// MI455X (gfx1250) — compile-verified
//
#include <hip/hip_runtime.h>
#include <hip/hip_bf16.h>

#define NINST 4096
#define BATCH 16
#define HH 1080
#define WW 1920

typedef float v2f __attribute__((ext_vector_type(2)));
typedef float v8f __attribute__((ext_vector_type(8)));

// One bilinear tap, exact 4-term form matching the reference arithmetic.
__device__ __forceinline__ float bilin_tap(const float* __restrict__ dmb,
                                           float cx, float cy, int s) {
    int k = s / 7;
    int j = s - k * 7;
    float gx = cx + (float)(j - 3) * (2.0f / (float)WW);
    float gy = cy + (float)(k - 3) * (2.0f / (float)HH);
    gx = fminf(fmaxf(gx, -1.0f), 1.0f);
    gy = fminf(fmaxf(gy, -1.0f), 1.0f);
    float ix = (gx + 1.0f) * (0.5f * (float)(WW - 1));
    float iy = (gy + 1.0f) * (0.5f * (float)(HH - 1));
    float x0 = floorf(ix);
    float y0 = floorf(iy);
    float wx = ix - x0;
    float wy = iy - y0;
    int x0i = min(max((int)x0, 0), WW - 1);
    int x1i = min(x0i + 1, WW - 1);
    int y0i = min(max((int)y0, 0), HH - 1);
    int y1i = min(y0i + 1, HH - 1);
    const float* r0 = dmb + (size_t)y0i * WW;
    const float* r1 = dmb + (size_t)y1i * WW;
    float d00 = r0[x0i];
    float d01 = r0[x1i];
    float d10 = r1[x0i];
    float d11 = r1[x1i];
    return d00 * (1.0f - wx) * (1.0f - wy)
         + d01 * wx          * (1.0f - wy)
         + d10 * (1.0f - wx) * wy
         + d11 * wx          * wy;
}

// One wave32 handles 16 instances. Lane m (low half) accumulates sample
// chunks 0,1 of instance m; lane m+16 accumulates chunks 2,3 — matching the
// V_WMMA_F32_16X16X4_F32 A-matrix layout (VGPR0 = K0/K2, VGPR1 = K1/K3).
// B = 1/49 (uniform) turns the WMMA into a per-instance 49-tap mean.
__global__ void __launch_bounds__(64)
depth_bbox_kernel(const float* __restrict__ bboxes,
                  const float* __restrict__ depth,
                  float* __restrict__ out) {
    const int lane  = threadIdx.x & 31;
    const int wave  = threadIdx.x >> 5;
    const int group = blockIdx.x * (blockDim.x >> 5) + wave; // 16 instances
    const int g     = group * 16;
    const int m     = lane & 15;
    const int hi    = lane >> 4;
    const int inst  = g + m;

    const float* bb = bboxes + (size_t)inst * 7;
    const float b0 = bb[0], b1 = bb[1], b2 = bb[2], b3 = bb[3];
    const float b4 = bb[4], b5 = bb[5], b6 = bb[6];

    const int bid = (int)b0;
    const float* __restrict__ dmb = depth + (size_t)bid * (size_t)HH * WW;

    const float cx = b3 + b5 - 1.0f;
    const float cy = b4 + b6 - 1.0f;

    // Warm L2 near the bbox center (gfx1250 global_prefetch path).
    {
        float gxc = fminf(fmaxf(cx, -1.0f), 1.0f);
        float gyc = fminf(fmaxf(cy, -1.0f), 1.0f);
        int icx = min(max((int)((gxc + 1.0f) * (0.5f * (float)(WW - 1))), 0), WW - 1);
        int icy = min(max((int)((gyc + 1.0f) * (0.5f * (float)(HH - 1))), 0), HH - 1);
        __builtin_prefetch(dmb + (size_t)icy * WW + icx, 0, 3);
    }

    // Chunks: c covers samples [13c, min(13c+13, 49)); sizes 13/13/13/10.
    const int s0 = (2 * hi) * 13;        // chunk 2*hi   (always 13 taps)
    const int s1 = s0 + 13;              // chunk 2*hi+1 (13 or 10 taps)
    const int e1 = min(s1 + 13, 49);

    float a0 = 0.0f;
    float a1 = 0.0f;
#pragma unroll 2
    for (int t = 0; t < 13; ++t) {
        a0 += bilin_tap(dmb, cx, cy, s0 + t);
        int ss = s1 + t;
        if (ss < e1) a1 += bilin_tap(dmb, cx, cy, ss);
    }

    // A: 16x4 f32 partial sums; B: uniform 1/49; D[m][*] = mean of instance m.
    v2f A;
    A[0] = a0;
    A[1] = a1;
    v2f Bm;
    Bm[0] = 1.0f / 49.0f;
    Bm[1] = 1.0f / 49.0f;
    v8f C = {};
    v8f D = __builtin_amdgcn_wmma_f32_16x16x4_f32(
        /*neg_a=*/false, A, /*neg_b=*/false, Bm,
        /*c_mod=*/(short)0, C, /*reuse_a=*/false, /*reuse_b=*/false);

    // D VGPR r: lanes 0-15 hold M=r, lanes 16-31 hold M=r+8.
    // Lanes 0..15 each write their own instance's output row (8 floats).
#pragma unroll
    for (int r = 0; r < 8; ++r) {
        float avg_hi = __shfl(D[r], 16, 32); // avg of instance g+8+r (from lane 16)
        if (lane == r) {
            float* o = out + (size_t)(g + r) * 8;
            o[0] = b0; o[1] = b1; o[2] = b2; o[3] = b3;
            o[4] = b4; o[5] = b5; o[6] = b6; o[7] = D[r];
        }
        if (lane == 8 + r) {
            float* o = out + (size_t)(g + 8 + r) * 8;
            o[0] = b0; o[1] = b1; o[2] = b2; o[3] = b3;
            o[4] = b4; o[5] = b5; o[6] = b6; o[7] = avg_hi;
        }
    }
}

extern "C" void kernel_launch(void* const* d_in, const int* in_sizes, int n_in,
                              void* d_out, int out_size, void* d_ws, size_t ws_size,
                              hipStream_t stream) {
    (void)in_sizes; (void)n_in; (void)out_size; (void)d_ws; (void)ws_size;
    const float* bboxes = (const float*)d_in[0];
    const float* depth  = (const float*)d_in[1];
    float* out = (float*)d_out;

    // 4096 instances / 16 per wave = 256 waves; 2 waves per 64-thread block.
    const int threads = 64;
    const int groups  = NINST / 16;                 // 256
    const int blocks  = groups / (threads / 32);    // 128
    depth_bbox_kernel<<<blocks, threads, 0, stream>>>(bboxes, depth, out);
}